// GIN_67765993996294
// MI455X (gfx1250) — compile-verified
//
#include <hip/hip_runtime.h>

typedef __attribute__((ext_vector_type(2))) float v2f;
typedef __attribute__((ext_vector_type(8))) float v8f;

#define N_IN  64
#define N_HID 128
#define N_OUTD 32
#define BN_EPS 1e-5f

// -------------------- row copy (float4) --------------------
__global__ void copy_f4(const float* __restrict__ src, float* __restrict__ dst,
                        long long n4) {
  long long i = (long long)blockIdx.x * blockDim.x + threadIdx.x;
  if (i < n4) ((float4*)dst)[i] = ((const float4*)src)[i];
}

// -------------------- edge scatter-add ---------------------
// z[dst] += h[src]; z pre-initialized with h (GIN eps=0).
// SHIFT: log2(dim/4). dim=64 -> 4, dim=128 -> 5.
template<int SHIFT>
__global__ void scatter_add(const float* __restrict__ h,
                            const int* __restrict__ srcIdx,
                            const int* __restrict__ dstIdx,
                            float* __restrict__ z, int nedges) {
  const int dim = (1 << SHIFT) * 4;
  long long tid = (long long)blockIdx.x * blockDim.x + threadIdx.x;
  int e = (int)(tid >> SHIFT);
  if (e >= nedges) return;
  int g = ((int)tid & ((1 << SHIFT) - 1)) * 4;
  int s = srcIdx[e], d = dstIdx[e];
  const float4 v = *(const float4*)&h[(long long)s * dim + g];
  float* p = &z[(long long)d * dim + g];
  unsafeAtomicAdd(p + 0, v.x);
  unsafeAtomicAdd(p + 1, v.y);
  unsafeAtomicAdd(p + 2, v.z);
  unsafeAtomicAdd(p + 3, v.w);
}

// -------------------- WMMA f32 GEMM ------------------------
// out[M x NOUT] = A[M x K] * W[K x NOUT] + bias  (optional ReLU)
// One wave computes 16 rows x NOUT via V_WMMA_F32_16X16X4_F32.
// W staged transposed in LDS ([col][k]) so each B fragment is one 8B load.
// Per k-step: batch-load all NT B fragments, then issue NT back-to-back WMMAs
// (avoids the serialized ds_load -> wait -> wmma pattern).
template<int K, int NOUT, bool RELU>
__global__ void gemm_wmma(const float* __restrict__ A, int lda,
                          const float* __restrict__ W,
                          const float* __restrict__ bias,
                          float* __restrict__ outp, int ldo, int ntiles) {
  extern __shared__ float lds[];              // NOUT*K floats (W^T)
  constexpr int NT = NOUT / 16;

  for (int idx = threadIdx.x; idx < K * NOUT; idx += blockDim.x) {
    int k = idx / NOUT, col = idx % NOUT;     // NOUT is power of 2
    lds[col * K + k] = W[idx];
  }
  __syncthreads();

  const int lane    = threadIdx.x & 31;
  const int laneLow = lane & 15;
  const int hiSel   = lane >> 4;              // 0: K/M base, 1: +2 / +8
  const int wavesPerBlock = blockDim.x >> 5;
  const int waveId  = blockIdx.x * wavesPerBlock + (threadIdx.x >> 5);
  const int nWaves  = gridDim.x * wavesPerBlock;

  // per-lane base into W^T (column block advances by 16*K floats per tile t)
  const float* ldsBase = lds + laneLow * K + 2 * hiSel;

  for (int tile = waveId; tile < ntiles; tile += nWaves) {
    const float* aRow = A + (long long)(tile * 16 + laneLow) * lda + 2 * hiSel;
    v8f acc[NT] = {};
#pragma unroll 2
    for (int k = 0; k < K; k += 4) {
      // A 16x4 f32 layout: VGPR0 = K (lanes0-15) / K+2 (lanes16-31); VGPR1 = +1
      v2f a = *(const v2f*)(aRow + k);
      // batch all B fragment loads so the compiler can issue them together
      v2f b[NT];
#pragma unroll
      for (int t = 0; t < NT; ++t)
        b[t] = *(const v2f*)(ldsBase + t * 16 * K + k);
#pragma unroll
      for (int t = 0; t < NT; ++t)
        acc[t] = __builtin_amdgcn_wmma_f32_16x16x4_f32(
            false, a, false, b[t], (short)0, acc[t], false, false);
    }
    // D layout: VGPR r -> M = r + 8*hiSel, N = laneLow
#pragma unroll
    for (int t = 0; t < NT; ++t) {
      int col = t * 16 + laneLow;
      float bb = bias[col];
      float* orow = outp + (long long)(tile * 16 + 8 * hiSel) * ldo + col;
#pragma unroll
      for (int r = 0; r < 8; ++r) {
        float v = acc[t][r] + bb;
        if (RELU) v = fmaxf(v, 0.0f);
        orow[(long long)r * ldo] = v;
      }
    }
  }
}

// -------------------- batchnorm ----------------------------
__global__ void bn_zero(float* __restrict__ stats) { stats[threadIdx.x] = 0.0f; }

#define BN_ROWS 200
__global__ void bn_stats(const float* __restrict__ c, float* __restrict__ stats,
                         int nrows) {
  int j = threadIdx.x;                        // 0..127 (one column per thread)
  int r0 = blockIdx.x * BN_ROWS;
  int r1 = min(r0 + BN_ROWS, nrows);
  float s = 0.f, s2 = 0.f;
  for (int r = r0; r < r1; ++r) {
    float v = c[(long long)r * N_HID + j];
    s += v; s2 += v * v;
  }
  unsafeAtomicAdd(&stats[j], s);
  unsafeAtomicAdd(&stats[N_HID + j], s2);
}

__global__ void bn_finalize(const float* __restrict__ stats,
                            const float* __restrict__ g,
                            const float* __restrict__ b,
                            float* __restrict__ ab, float invN) {
  int j = threadIdx.x;                        // 128 threads
  float m   = stats[j] * invN;
  float var = stats[N_HID + j] * invN - m * m;
  float a   = g[j] * rsqrtf(var + BN_EPS);
  ab[j]         = a;
  ab[N_HID + j] = b[j] - m * a;
}

// BN + skip + ReLU, save pre-relu residual, h written in place over c
__global__ void fuse_bn_skip_relu(float* __restrict__ c,
                                  const float* __restrict__ skip,
                                  const float* __restrict__ ab,
                                  float* __restrict__ res, long long n) {
  long long i = (long long)blockIdx.x * blockDim.x + threadIdx.x;
  if (i >= n) return;
  int j = (int)(i & (N_HID - 1));
  float v = ab[j] * c[i] + ab[N_HID + j] + skip[i];
  res[i] = v;
  c[i]   = fmaxf(v, 0.0f);
}

// BN + residual + ReLU, h in place over c
__global__ void fuse_bn_res_relu(float* __restrict__ c,
                                 const float* __restrict__ res,
                                 const float* __restrict__ ab, long long n) {
  long long i = (long long)blockIdx.x * blockDim.x + threadIdx.x;
  if (i >= n) return;
  int j = (int)(i & (N_HID - 1));
  c[i] = fmaxf(ab[j] * c[i] + ab[N_HID + j] + res[i], 0.0f);
}

// -------------------- launch -------------------------------
extern "C" void kernel_launch(void* const* d_in, const int* in_sizes, int n_in,
                              void* d_out, int out_size, void* d_ws, size_t ws_size,
                              hipStream_t stream) {
  const float* x    = (const float*)d_in[0];
  const int*   eidx = (const int*)d_in[1];
  const float* w1_0 = (const float*)d_in[2];  const float* b1_0 = (const float*)d_in[3];
  const float* w2_0 = (const float*)d_in[4];  const float* b2_0 = (const float*)d_in[5];
  const float* skw  = (const float*)d_in[6];  const float* skb  = (const float*)d_in[7];
  const float* g0   = (const float*)d_in[8];  const float* bb0  = (const float*)d_in[9];
  const float* w1_1 = (const float*)d_in[10]; const float* b1_1 = (const float*)d_in[11];
  const float* w2_1 = (const float*)d_in[12]; const float* b2_1 = (const float*)d_in[13];
  const float* g1   = (const float*)d_in[14]; const float* bb1  = (const float*)d_in[15];
  const float* w1_2 = (const float*)d_in[16]; const float* b1_2 = (const float*)d_in[17];
  const float* w2_2 = (const float*)d_in[18]; const float* b2_2 = (const float*)d_in[19];

  const int nnodes = in_sizes[0] / N_IN;      // 50000
  const int nedges = in_sizes[1] / 2;         // 800000
  const int* srcI = eidx;
  const int* dstI = eidx + nedges;

  const size_t be = (size_t)nnodes * N_HID;
  float* B0 = (float*)d_ws;                   // z (aggregated input)
  float* B1 = B0 + be;                        // t = relu(z@W1+b1)
  float* B2 = B1 + be;                        // c = t@W2+b2, then h (in place)
  float* B3 = B2 + be;                        // skip projection (layer 0)
  float* B4 = B3 + be;                        // residual
  float* stats = B4 + be;                     // [sum(128), sumsq(128)]
  float* ab    = stats + 2 * N_HID;           // [scale(128), shift(128)]

  const int ntiles = nnodes / 16;             // 3125 (exact)
  const long long nh = (long long)nnodes * N_HID;
  const int GB = 160, GT = 256;               // gemm grid/block
  const int bnBlocks = (nnodes + BN_ROWS - 1) / BN_ROWS;
  const float invN = 1.0f / (float)nnodes;

  // scatter grids
  const long long s64T  = (long long)nedges * 16;
  const long long s128T = (long long)nedges * 32;
  const int s64B  = (int)((s64T  + 255) / 256);
  const int s128B = (int)((s128T + 255) / 256);
  const int cp64B  = (int)(((long long)nnodes * 16 + 255) / 256);
  const int cp128B = (int)(((long long)nnodes * 32 + 255) / 256);
  const int fuseB  = (int)((nh + 255) / 256);

  // ---------------- layer 0 ----------------
  copy_f4<<<cp64B, 256, 0, stream>>>(x, B0, (long long)nnodes * 16);
  scatter_add<4><<<s64B, 256, 0, stream>>>(x, srcI, dstI, B0, nedges);
  gemm_wmma<64, 128, true ><<<GB, GT, 64 * 128 * 4, stream>>>(B0, 64, w1_0, b1_0, B1, 128, ntiles);
  gemm_wmma<128, 128, false><<<GB, GT, 128 * 128 * 4, stream>>>(B1, 128, w2_0, b2_0, B2, 128, ntiles);
  gemm_wmma<64, 128, false><<<GB, GT, 64 * 128 * 4, stream>>>(x, 64, skw, skb, B3, 128, ntiles);
  bn_zero<<<1, 256, 0, stream>>>(stats);
  bn_stats<<<bnBlocks, 128, 0, stream>>>(B2, stats, nnodes);
  bn_finalize<<<1, 128, 0, stream>>>(stats, g0, bb0, ab, invN);
  fuse_bn_skip_relu<<<fuseB, 256, 0, stream>>>(B2, B3, ab, B4, nh);  // B2 := h

  // ---------------- layer 1 ----------------
  copy_f4<<<cp128B, 256, 0, stream>>>(B2, B0, (long long)nnodes * 32);
  scatter_add<5><<<s128B, 256, 0, stream>>>(B2, srcI, dstI, B0, nedges);
  gemm_wmma<128, 128, true ><<<GB, GT, 128 * 128 * 4, stream>>>(B0, 128, w1_1, b1_1, B1, 128, ntiles);
  gemm_wmma<128, 128, false><<<GB, GT, 128 * 128 * 4, stream>>>(B1, 128, w2_1, b2_1, B2, 128, ntiles);
  bn_zero<<<1, 256, 0, stream>>>(stats);
  bn_stats<<<bnBlocks, 128, 0, stream>>>(B2, stats, nnodes);
  bn_finalize<<<1, 128, 0, stream>>>(stats, g1, bb1, ab, invN);
  fuse_bn_res_relu<<<fuseB, 256, 0, stream>>>(B2, B4, ab, nh);       // B2 := h

  // ---------------- layer 2 ----------------
  copy_f4<<<cp128B, 256, 0, stream>>>(B2, B0, (long long)nnodes * 32);
  scatter_add<5><<<s128B, 256, 0, stream>>>(B2, srcI, dstI, B0, nedges);
  gemm_wmma<128, 128, true ><<<GB, GT, 128 * 128 * 4, stream>>>(B0, 128, w1_2, b1_2, B1, 128, ntiles);
  gemm_wmma<128, 32, false><<<GB, GT, 128 * 32 * 4, stream>>>(B1, 128, w2_2, b2_2,
                                                              (float*)d_out, 32, ntiles);
}